// LossWithLS_1047972020265
// MI455X (gfx1250) — compile-verified
//
#include <hip/hip_runtime.h>
#include <math.h>

// Label-smoothed KLDiv loss, MI455X (gfx1250).
//
// Roofline: 8192x32000 fp32 read once = 1.05 GB @ 23.3 TB/s => ~45us memory
// bound. Row sums are done as D = ones(16x4) x B(4x16) + C via
// V_WMMA_F32_16X16X4_F32: with A==1, D[m][n] = sum_k B[k][n] + C[m][n].
// Since we finally sum the entire D tile, element placement inside B is
// irrelevant -> feed B straight from coalesced b128 loads, no shuffles.

typedef __attribute__((ext_vector_type(2))) float v2f;
typedef __attribute__((ext_vector_type(4))) float v4f;
typedef __attribute__((ext_vector_type(8))) float v8f;

#define VOCAB 32000
#define NROWS 8192          // 4 * 2048
#define WAVES_PER_BLOCK 8   // 256 threads, wave32
#define SMOOTH_F 0.1f
#define CONF_F 0.9f

__global__ __launch_bounds__(256) void
rowloss_kernel(const float* __restrict__ pred,
               const int* __restrict__ target,
               const int* __restrict__ mask,
               float* __restrict__ row_loss) {
  const int lane = threadIdx.x & 31;
  const int wave = threadIdx.x >> 5;
  const int row  = blockIdx.x * WAVES_PER_BLOCK + wave;
  const float* rp = pred + (size_t)row * VOCAB;   // 128B-aligned (32000*4)

  v8f acc0 = {};
  v8f acc1 = {};
  v2f a_ones;
  a_ones.x = 1.0f;
  a_ones.y = 1.0f;

  // 32000 = 250 * 128 floats; per iter: one b128 load (512B/wave, coalesced,
  // non-temporal: single-pass 1GB stream, don't thrash 192MB L2) feeding two
  // v_wmma_f32_16x16x4_f32 accumulations. Unroll 5 => 5 loads in flight.
#pragma unroll 5
  for (int it = 0; it < 250; ++it) {
    const v4f d = __builtin_nontemporal_load(
        (const v4f*)(rp + (size_t)it * 128 + lane * 4));
    v2f b0; b0.x = d.x; b0.y = d.y;
    v2f b1; b1.x = d.z; b1.y = d.w;
    acc0 = __builtin_amdgcn_wmma_f32_16x16x4_f32(
        /*neg_a=*/false, a_ones, /*neg_b=*/false, b0,
        /*c_mod=*/(short)0, acc0, /*reuse_a=*/false, /*reuse_b=*/false);
    acc1 = __builtin_amdgcn_wmma_f32_16x16x4_f32(
        false, a_ones, false, b1, (short)0, acc1, false, false);
  }

  // D[m][n] is independent of m (A==1): every VGPR of the accumulator holds
  // that lane's column sum; lanes 16-31 duplicate columns 0-15. Reduce over
  // all 32 lanes and halve -> exact row sum.
  float tot = acc0[0] + acc1[0];
#pragma unroll
  for (int off = 16; off > 0; off >>= 1)
    tot += __shfl_xor(tot, off, 32);
  tot *= 0.5f;

  if (lane == 0) {
    const float sv = SMOOTH_F / (float)(VOCAB - 1);
    const float ent =
        (float)(VOCAB - 1) * sv * logf(sv) + CONF_F * logf(CONF_F);
    const int   tgt = target[row];
    const float pt  = rp[tgt];
    const float m   = (float)mask[row];
    const float loss = ent - (sv * tot + (CONF_F - sv) * pt);
    row_loss[row] = loss * m;
  }
}

// Deterministic final reduction: fixed-order grid-stride accumulation +
// shared-memory tree; integer mask sum. No float atomics -> bit-stable
// across graph replays.
__global__ __launch_bounds__(256) void
finalize_kernel(const float* __restrict__ row_loss,
                const int* __restrict__ mask,
                float* __restrict__ out) {
  __shared__ float sl[256];
  __shared__ int   sm[256];
  const int tid = threadIdx.x;
  float ls = 0.0f;
  int   ms = 0;
  for (int i = tid; i < NROWS; i += 256) {
    ls += row_loss[i];
    ms += mask[i];
  }
  sl[tid] = ls;
  sm[tid] = ms;
  __syncthreads();
#pragma unroll
  for (int s = 128; s > 0; s >>= 1) {
    if (tid < s) {
      sl[tid] += sl[tid + s];
      sm[tid] += sm[tid + s];
    }
    __syncthreads();
  }
  if (tid == 0) out[0] = sl[0] / (float)sm[0];
}

extern "C" void kernel_launch(void* const* d_in, const int* in_sizes, int n_in,
                              void* d_out, int out_size, void* d_ws,
                              size_t ws_size, hipStream_t stream) {
  const float* pred   = (const float*)d_in[0];  // (4,2048,32000) f32
  const int*   target = (const int*)d_in[1];    // (4,2048) i32
  const int*   mask   = (const int*)d_in[2];    // (4,2048) i32
  float*       out    = (float*)d_out;          // scalar f32
  float* row_loss = (float*)d_ws;               // 8192 floats = 32KB scratch

  rowloss_kernel<<<NROWS / WAVES_PER_BLOCK, 256, 0, stream>>>(pred, target,
                                                              mask, row_loss);
  finalize_kernel<<<1, 256, 0, stream>>>(row_loss, mask, out);
}